// SparseImageWarp_28793460752583
// MI455X (gfx1250) — compile-verified
//
#include <hip/hip_runtime.h>
#include <math.h>

typedef __attribute__((ext_vector_type(2))) float v2f;
typedef __attribute__((ext_vector_type(8))) float v8f;

#define WN 8
#define WC 3
#define WH 512
#define WW 512
#define NCOEF 20                 // 16 phi weights + x,y,1 affine + 1 zero pad
#define EPS2 1.42108547152020037174e-14f   // (float eps)^2

// ---------------------------------------------------------------------------
// Kernel 1: per-batch 19x19 TPS solve (Gauss-Jordan w/ partial pivoting).
// One wave (32 threads) per batch. Result: coef[n][20][2] in d_ws.
// coef rows 0..15 = phi weights, 16..18 = affine (x,y,const), 19 = zero pad.
// ---------------------------------------------------------------------------
__global__ void tps_solve_kernel(const float* __restrict__ src,
                                 const float* __restrict__ dst,
                                 float* __restrict__ coef)
{
    const int n = blockIdx.x;
    const int t = threadIdx.x;               // 0..31
    __shared__ float aug[19][21];            // 19 lhs cols + 2 rhs cols
    __shared__ float cx[16], cy[16], fx[16], fy[16];
    __shared__ int s_piv;

    if (t < 16) {
        // reference flips last dim: c = dest[..., ::-1] -> (x, y)
        float dx = dst[(n*16 + t)*2 + 1];
        float dy = dst[(n*16 + t)*2 + 0];
        cx[t] = dx; cy[t] = dy;
        fx[t] = dx - src[(n*16 + t)*2 + 1];  // f = dst - src (x component)
        fy[t] = dy - src[(n*16 + t)*2 + 0];  // (y component)
    }
    __syncthreads();

    if (t < 19) {
        if (t < 16) {
            for (int j = 0; j < 16; ++j) {
                float dx = cx[t] - cx[j], dy = cy[t] - cy[j];
                float r2 = dx*dx + dy*dy;
                // phi(r) = r^2 log(max(r,eps)) = 0.5 r^2 log(max(r^2, eps^2))
                aug[t][j] = 0.5f * r2 * logf(fmaxf(r2, EPS2));
            }
            aug[t][16] = cx[t]; aug[t][17] = cy[t]; aug[t][18] = 1.0f;
            aug[t][19] = fx[t]; aug[t][20] = fy[t];
        } else {
            for (int j = 0; j < 16; ++j)
                aug[t][j] = (t == 16) ? cx[j] : (t == 17) ? cy[j] : 1.0f;
            for (int j = 16; j < 21; ++j) aug[t][j] = 0.0f;
        }
    }
    __syncthreads();

    for (int k = 0; k < 19; ++k) {
        if (t == 0) {
            int p = k; float best = fabsf(aug[k][k]);
            for (int i = k + 1; i < 19; ++i) {
                float v = fabsf(aug[i][k]);
                if (v > best) { best = v; p = i; }
            }
            s_piv = p;
        }
        __syncthreads();
        const int p = s_piv;
        if (p != k && t < 21) {
            float tmp = aug[k][t]; aug[k][t] = aug[p][t]; aug[p][t] = tmp;
        }
        __syncthreads();
        if (t < 19 && t != k) {
            float f = aug[t][k] / aug[k][k];
            for (int j = k; j < 21; ++j)
                aug[t][j] -= f * aug[k][j];
        }
        __syncthreads();
    }

    if (t < 19) {
        float inv = 1.0f / aug[t][t];
        coef[n*2*NCOEF + t*2 + 0] = aug[t][19] * inv;   // x-flow coeffs
        coef[n*2*NCOEF + t*2 + 1] = aug[t][20] * inv;   // y-flow coeffs
    } else if (t == 19) {
        coef[n*2*NCOEF + 38] = 0.0f;                    // K-pad row
        coef[n*2*NCOEF + 39] = 0.0f;
    }
}

// ---------------------------------------------------------------------------
// Feature j of pixel (x,y): j<16 -> phi(dist to center j); 16->x; 17->y;
// 18->1; 19->0 (K padding).
// ---------------------------------------------------------------------------
__device__ __forceinline__ float tps_feat(int j, float x, float y,
                                          const float* cx, const float* cy)
{
    if (j < 16) {
        float dx = x - cx[j], dy = y - cy[j];
        float r2 = dx*dx + dy*dy;
        return 0.5f * r2 * logf(fmaxf(r2, EPS2));
    }
    if (j == 16) return x;
    if (j == 17) return y;
    if (j == 18) return 1.0f;
    return 0.0f;
}

// ---------------------------------------------------------------------------
// Kernel 2: per-pixel flow via V_WMMA_F32_16X16X4_F32 + bilinear warp.
// Each wave owns 32 row-contiguous pixels. A = coef^T (rows 0/1 = flow x/y),
// B = 20 features x 16 pixels, split into 5 K=4 chunks; two accumulators
// cover the two 16-pixel half-tiles (10 WMMAs / wave).
// A frag (16x4 f32): lane=M, V0=K0|K2, V1=K1|K3.
// B frag (4x16 f32): lane=N, V0=K0|K2, V1=K1|K3.
// D frag: V0 = row M=0 (flow_x) lanes 0-15, V1 = row M=1 (flow_y).
// ---------------------------------------------------------------------------
__global__ void tps_warp_kernel(const float* __restrict__ image,
                                const float* __restrict__ dst,
                                const float* __restrict__ coef,
                                float* __restrict__ warped,
                                float* __restrict__ flow_out)
{
    const int n = blockIdx.y;
    __shared__ float s_coef[NCOEF][2];
    __shared__ float s_cx[16], s_cy[16];

    if (threadIdx.x < 2*NCOEF)
        ((float*)s_coef)[threadIdx.x] = coef[n*2*NCOEF + threadIdx.x];
    if (threadIdx.x >= 64 && threadIdx.x < 80) {
        int mm = threadIdx.x - 64;
        s_cx[mm] = dst[(n*16 + mm)*2 + 1];
        s_cy[mm] = dst[(n*16 + mm)*2 + 0];
    }
    __syncthreads();

    const int  lane = threadIdx.x & 31;
    const int  wv   = threadIdx.x >> 5;
    const int  m    = lane & 15;
    const bool hi   = lane >= 16;
    const int  pixBase = blockIdx.x * 256 + wv * 32;   // 8 waves x 32 px

    // pixels whose B columns this lane supplies (column index = m)
    const int pA = pixBase + m;          // half-tile 0: pixels 0..15
    const int pB = pixBase + 16 + m;     // half-tile 1: pixels 16..31
    const float xA = (float)(pA & (WW-1)), yA = (float)(pA >> 9);
    const float xB = (float)(pB & (WW-1)), yB = (float)(pB >> 9);

    v2f a[5], b1[5], b2[5];
    #pragma unroll
    for (int k = 0; k < 5; ++k) {
        const int j0 = 4*k + (hi ? 2 : 0);   // K indices this lane holds
        const int j1 = j0 + 1;
        a[k].x = (m < 2) ? s_coef[j0][m] : 0.0f;   // row m of coef^T
        a[k].y = (m < 2) ? s_coef[j1][m] : 0.0f;
        b1[k].x = tps_feat(j0, xA, yA, s_cx, s_cy);
        b1[k].y = tps_feat(j1, xA, yA, s_cx, s_cy);
        b2[k].x = tps_feat(j0, xB, yB, s_cx, s_cy);
        b2[k].y = tps_feat(j1, xB, yB, s_cx, s_cy);
    }

    v8f acc1 = {};
    v8f acc2 = {};
    #pragma unroll
    for (int k = 0; k < 5; ++k) {
        acc1 = __builtin_amdgcn_wmma_f32_16x16x4_f32(
            false, a[k], false, b1[k], (short)0, acc1, false, false);
        acc2 = __builtin_amdgcn_wmma_f32_16x16x4_f32(
            false, a[k], false, b2[k], (short)0, acc2, false, false);
    }

    // D row0/row1 hold flow_x/flow_y for column = pixel; gather own pixel.
    const float fx1 = __shfl(acc1[0], m, 32);
    const float fy1 = __shfl(acc1[1], m, 32);
    const float fx2 = __shfl(acc2[0], m, 32);
    const float fy2 = __shfl(acc2[1], m, 32);
    const float fx = hi ? fx2 : fx1;
    const float fy = hi ? fy2 : fy1;

    const int pix = pixBase + lane;
    const int px = pix & (WW-1), py = pix >> 9;

    // grid->sample coordinate algebra collapses to (p - flow), border clamp
    const float xs = fminf(fmaxf((float)px - fx, 0.0f), (float)(WW-1));
    const float ys = fminf(fmaxf((float)py - fy, 0.0f), (float)(WH-1));
    const float x0f = floorf(xs), y0f = floorf(ys);
    const float wx = xs - x0f,  wy = ys - y0f;
    int x0 = min(max((int)x0f, 0), WW-1);
    int y0 = min(max((int)y0f, 0), WH-1);
    int x1 = min(x0 + 1, WW-1);
    int y1 = min(y0 + 1, WH-1);

    const float w00 = (1.0f-wx)*(1.0f-wy), w01 = wx*(1.0f-wy);
    const float w10 = (1.0f-wx)*wy,        w11 = wx*wy;

    #pragma unroll
    for (int c = 0; c < WC; ++c) {
        const float* img = image + (size_t)(n*WC + c) * (WH*WW);
        const float Ia = img[y0*WW + x0];
        const float Ib = img[y0*WW + x1];
        const float Ic = img[y1*WW + x0];
        const float Id = img[y1*WW + x1];
        warped[((size_t)(n*WC + c)*WH + py)*WW + px] =
            Ia*w00 + Ib*w01 + Ic*w10 + Id*w11;
    }
    // flow_hw = flow[..., ::-1]  -> store (fy, fx)
    const size_t fo = ((size_t)(n*WH + py)*WW + px) * 2;
    flow_out[fo + 0] = fy;
    flow_out[fo + 1] = fx;
}

extern "C" void kernel_launch(void* const* d_in, const int* in_sizes, int n_in,
                              void* d_out, int out_size, void* d_ws, size_t ws_size,
                              hipStream_t stream)
{
    const float* image = (const float*)d_in[0];
    const float* srcp  = (const float*)d_in[1];
    const float* dstp  = (const float*)d_in[2];
    float* warped = (float*)d_out;
    float* flow   = warped + (size_t)WN * WC * WH * WW;
    float* coef   = (float*)d_ws;                      // 8 * 40 floats

    tps_solve_kernel<<<dim3(WN), dim3(32), 0, stream>>>(srcp, dstp, coef);
    tps_warp_kernel<<<dim3((WH*WW)/256, WN), dim3(256), 0, stream>>>(
        image, dstp, coef, warped, flow);
}